// WavetableOsc_86732569576149
// MI455X (gfx1250) — compile-verified
//
#include <hip/hip_runtime.h>
#include <math.h>

// ---------------- problem constants (from reference) ----------------
#define SRATE   48000.0f
#define NS      131072          // samples per batch row
#define BS      32              // batch size
#define NPOS    256             // wavetable positions
#define NWT     2048            // wavetable length
#define AAN     257             // AA filter taps  (NWT/8 + 1)
#define KPAD    260             // taps padded to multiple of 4 for WMMA K-steps
#define NPADC   2304            // NWT + 2*128 circular pad
#define PADW    2308            // NPADC + 4 zero tail (so A-frag reads never OOB)
#define SEGLEN  128             // samples per scan segment
#define NSEG    1024            // NS / SEGLEN
#define MTOT    (NPOS * NWT)    // 524288 output rows of the conv-GEMM

#define TWO_PI  6.28318530717958647692f

typedef float v2f __attribute__((ext_vector_type(2)));
typedef float v8f __attribute__((ext_vector_type(8)));

// ---------------- helpers ----------------
__device__ __forceinline__ float wave_incl_scan(float v, int lane) {
    #pragma unroll
    for (int off = 1; off < 32; off <<= 1) {
        float n = __shfl_up(v, off, 32);
        if (lane >= off) v += n;
    }
    return v;
}

__device__ __forceinline__ float wave_max(float v) {
    #pragma unroll
    for (int off = 16; off; off >>= 1)
        v = fmaxf(v, __shfl_xor(v, off, 32));
    return v;
}

// ---------------- K1: per-row segmented scan of 2*pi*f0/SR + row max -> cf ----------------
// grid = 32 blocks (one per batch row), 1024 threads (32 waves), SEGLEN=128 elems/thread
__global__ void wto_scan_max_kernel(const float* __restrict__ f0,
                                    float* __restrict__ seg_off,  // [BS][NSEG] exclusive offsets
                                    float* __restrict__ cf) {     // [BS] cutoff Hz
    const int r    = blockIdx.x;
    const int t    = threadIdx.x;
    const int lane = t & 31;
    const int wid  = t >> 5;
    const float k  = TWO_PI / SRATE;
    const float* row = f0 + (size_t)r * NS;

    float s = 0.f, m = 0.f;
    const int base = t * SEGLEN;
    #pragma unroll 4
    for (int i = 0; i < SEGLEN; ++i) {
        float v = row[base + i];
        s += v * k;
        m = fmaxf(m, v);
    }
    const float own = s;
    float incl = wave_incl_scan(s, lane);
    m = wave_max(m);

    __shared__ float wsum[32], wmx[32];
    if (lane == 31) wsum[wid] = incl;
    if (lane == 0)  wmx[wid]  = m;
    __syncthreads();
    if (wid == 0) {
        float wi = wave_incl_scan(wsum[lane], lane);
        wsum[lane] = wi;
        float mm = wave_max(wmx[lane]);
        if (lane == 0) {
            // cf = SR / (max_f0 / wt_pitch) / 2 * 0.9 ;  wt_pitch = SR/NWT
            cf[r] = SRATE * (SRATE / (float)NWT) / mm * 0.45f;
        }
    }
    __syncthreads();
    const float wexcl = (wid > 0) ? wsum[wid - 1] : 0.f;
    seg_off[r * NSEG + t] = wexcl + (incl - own);   // exclusive prefix for this segment
}

// ---------------- K2: build normalized AA filters, store transposed hT[k][b] ----------------
// grid = 32 blocks (one per batch), 512 threads
__global__ void wto_filter_kernel(const float* __restrict__ cf,
                                  float* __restrict__ hT) {        // [KPAD][BS]
    const int b = blockIdx.x;
    const int i = threadIdx.x;
    __shared__ float red[512];

    float hv = 0.f;
    if (i < AAN) {
        const float support = 2.f * ((float)i - (float)(AAN - 1) * 0.5f) / SRATE;
        const float x  = cf[b] * support;
        const float px = (float)M_PI * x;
        const float snc = (fabsf(x) < 1e-9f) ? 1.f : (sinf(px) / px);
        const float mm = (float)i / (float)(AAN - 1);
        const float bw = 0.42f - 0.5f * cosf(TWO_PI * mm) + 0.08f * cosf(2.f * TWO_PI * mm);
        hv = snc * bw;
    }
    red[i] = hv;
    __syncthreads();
    #pragma unroll
    for (int off = 256; off; off >>= 1) {
        if (i < off) red[i] += red[i + off];
        __syncthreads();
    }
    const float inv = 1.f / red[0];
    if (i < AAN)        hT[i * BS + b] = hv * inv;
    else if (i < KPAD)  hT[i * BS + b] = 0.f;       // K-padding taps
}

// ---------------- K3: tanh + circular pad : padded[p][0..2307] ----------------
__global__ void wto_tanh_pad_kernel(const float* __restrict__ wt,
                                    float* __restrict__ padded) {
    const int idx = blockIdx.x * blockDim.x + threadIdx.x;
    if (idx >= NPOS * PADW) return;
    const int p = idx / PADW;
    const int j = idx - p * PADW;
    float v = 0.f;
    if (j < NPADC) {
        const int src = (j + NWT - 128) & (NWT - 1);   // (j-128) mod 2048
        v = tanhf(wt[p * NWT + src]);
    }
    padded[idx] = v;
}

// ---------------- K4: depthwise conv as GEMM via V_WMMA_F32_16X16X4_F32 ----------------
// filt[b][m] = sum_k padded[p][x+k] * h[b][k],  m = p*NWT + x
// One wave per 16-row m-tile; wave computes 16(M) x 32(N=batch) via two accumulators.
// A-frag layout (ISA 7.12.2, 32-bit A 16x4): lanes 0-15 -> K = k0+{0,1}; lanes 16-31 -> K = k0+{2,3}
// B-frag assumed mirrored: lanes 0-15 -> rows k0+{0,1}; lanes 16-31 -> rows k0+{2,3}, N = lane&15
// D layout: VGPR i, lane l: M = i + 8*(l>>4), N = l&15
__global__ void wto_conv_wmma_kernel(const float* __restrict__ padded,
                                     const float* __restrict__ hT,
                                     float* __restrict__ filt) {
    const int lane   = threadIdx.x & 31;
    const int wid    = blockIdx.x * (blockDim.x >> 5) + (threadIdx.x >> 5);
    const int m_base = wid << 4;                 // 16-row tile, never crosses a p row
    const int p      = m_base >> 11;             // /NWT
    const int x      = (m_base & (NWT - 1)) + (lane & 15);
    const int koff   = (lane >> 4) << 1;         // 0 or 2
    const int nb     = lane & 15;

    const float* aptr = padded + (size_t)p * PADW + x + koff;

    v8f acc0 = {};   // batches 0..15
    v8f acc1 = {};   // batches 16..31

    for (int k0 = 0; k0 < KPAD; k0 += 4) {
        v2f a, b0, b1;
        a.x = aptr[k0];
        a.y = aptr[k0 + 1];
        const float* hrow = hT + (k0 + koff) * BS + nb;
        b0.x = hrow[0];
        b0.y = hrow[BS];
        b1.x = hrow[16];
        b1.y = hrow[BS + 16];
        acc0 = __builtin_amdgcn_wmma_f32_16x16x4_f32(false, a, false, b0,
                                                     (short)0, acc0, false, false);
        acc1 = __builtin_amdgcn_wmma_f32_16x16x4_f32(false, a, false, b1,
                                                     (short)0, acc1, false, false);
    }

    const int mrow = m_base + ((lane >> 4) << 3);          // + 0 or + 8
    float* d0 = filt + (size_t)nb * MTOT + mrow;           // batch nb
    float* d1 = filt + (size_t)(nb + 16) * MTOT + mrow;    // batch nb+16
    float4 t;
    t.x = acc0[0]; t.y = acc0[1]; t.z = acc0[2]; t.w = acc0[3];
    *(float4*)(d0) = t;
    t.x = acc0[4]; t.y = acc0[5]; t.z = acc0[6]; t.w = acc0[7];
    *(float4*)(d0 + 4) = t;
    t.x = acc1[0]; t.y = acc1[1]; t.z = acc1[2]; t.w = acc1[3];
    *(float4*)(d1) = t;
    t.x = acc1[4]; t.y = acc1[5]; t.z = acc1[6]; t.w = acc1[7];
    *(float4*)(d1 + 4) = t;
}

// ---------------- K5: per-segment sequential phase + bilinear gather ----------------
// 32768 threads, one 128-sample segment each; filt is L2-resident (67MB < 192MB L2)
__global__ void wto_synth_kernel(const float* __restrict__ f0,
                                 const float* __restrict__ wpos,
                                 const float* __restrict__ phase,
                                 const float* __restrict__ seg_off,
                                 const float* __restrict__ filt,
                                 float* __restrict__ out) {
    const int g = blockIdx.x * blockDim.x + threadIdx.x;   // 0 .. BS*NSEG-1
    const int r = g >> 10;                                  // / NSEG
    const int s = g & (NSEG - 1);
    const float k  = TWO_PI / SRATE;
    float run      = seg_off[g];
    const float ph = phase[r];

    const size_t base = (size_t)r * NS + (size_t)s * SEGLEN;
    const float* frow = f0   + base;
    const float* wrow = wpos + base;
    const float* fb   = filt + (size_t)r * MTOT;
    float*       orow = out  + base;

    for (int i = 0; i < SEGLEN; ++i) {
        run += k * frow[i];                   // inclusive prefix
        float a = fmodf(run + ph, TWO_PI);
        if (a < 0.f) a += TWO_PI;             // python-style mod -> [0, 2pi)
        float ix = a * ((float)(NWT - 1) / TWO_PI);
        ix = fminf(fmaxf(ix, 0.f), (float)(NWT - 1));
        float iy = fminf(fmaxf(wrow[i] * (float)(NPOS - 1), 0.f), (float)(NPOS - 1));

        const float x0 = floorf(ix), y0 = floorf(iy);
        const int x0i = (int)x0, y0i = (int)y0;
        const int x1i = min(x0i + 1, NWT - 1);
        const int y1i = min(y0i + 1, NPOS - 1);
        const float wx = ix - x0, wy = iy - y0;

        const float* ry0 = fb + (size_t)y0i * NWT;
        const float* ry1 = fb + (size_t)y1i * NWT;
        const float v00 = ry0[x0i], v01 = ry0[x1i];
        const float v10 = ry1[x0i], v11 = ry1[x1i];
        orow[i] = (v00 * (1.f - wx) + v01 * wx) * (1.f - wy)
                + (v10 * (1.f - wx) + v11 * wx) * wy;
    }
}

// ---------------- launcher ----------------
extern "C" void kernel_launch(void* const* d_in, const int* in_sizes, int n_in,
                              void* d_out, int out_size, void* d_ws, size_t ws_size,
                              hipStream_t stream) {
    (void)in_sizes; (void)n_in; (void)out_size; (void)ws_size;
    const float* f0    = (const float*)d_in[0];   // (32, 131072)
    const float* wpos  = (const float*)d_in[1];   // (32, 131072)
    const float* phase = (const float*)d_in[2];   // (32,)
    const float* wt    = (const float*)d_in[3];   // (256, 2048)
    float* out = (float*)d_out;

    // workspace layout (floats, 64-float aligned slabs)
    float* ws     = (float*)d_ws;
    float* cf     = ws;                     // 32
    float* seg    = ws + 64;                // 32768
    float* hT     = seg + 32768;            // 260*32 = 8320 (slab 8384)
    float* padded = hT + 8384;              // 256*2308 = 590848
    float* filt   = padded + 590848;        // 32*524288 = 16777216  (~67MB)

    wto_scan_max_kernel<<<BS, 1024, 0, stream>>>(f0, seg, cf);
    wto_filter_kernel<<<BS, 512, 0, stream>>>(cf, hT);
    wto_tanh_pad_kernel<<<(NPOS * PADW + 255) / 256, 256, 0, stream>>>(wt, padded);
    // MTOT/16 = 32768 waves, 8 waves (256 threads) per block -> 4096 blocks
    wto_conv_wmma_kernel<<<4096, 256, 0, stream>>>(padded, hT, filt);
    wto_synth_kernel<<<(BS * NSEG) / 256, 256, 0, stream>>>(f0, wpos, phase, seg, filt, out);
}